// GCN_Net_16243566313846
// MI455X (gfx1250) — compile-verified
//
#include <hip/hip_runtime.h>
#include <hip/hip_bf16.h>

typedef float v2f __attribute__((ext_vector_type(2)));
typedef float v8f __attribute__((ext_vector_type(8)));

#define F_HID 64

static __device__ __forceinline__ v8f wmma_f32_k4(v2f a, v2f b, v8f c) {
    // D = A(16x4,f32) * B(4x16,f32) + C(16x16,f32)
    return __builtin_amdgcn_wmma_f32_16x16x4_f32(false, a, false, b, (short)0, c, false, false);
}

static __device__ __forceinline__ void atomic_add_f32(float* p, float v) {
    __hip_atomic_fetch_add(p, v, __ATOMIC_RELAXED, __HIP_MEMORY_SCOPE_AGENT);
}

// ---------------- degree / norm ----------------

__global__ void deg_init_kernel(float* __restrict__ deg, int n) {
    int t = blockIdx.x * blockDim.x + threadIdx.x;
    if (t < n) deg[t] = 1.0f;  // self loop contributes 1
}

__global__ void deg_accum_kernel(const int* __restrict__ dst, float* __restrict__ deg, int e) {
    int t = blockIdx.x * blockDim.x + threadIdx.x;
    if (t < e) atomic_add_f32(deg + dst[t], 1.0f);
}

__global__ void rsqrt_kernel(float* __restrict__ deg, int n) {
    int t = blockIdx.x * blockDim.x + threadIdx.x;
    if (t < n) deg[t] = rsqrtf(fmaxf(deg[t], 1e-12f));
}

__global__ void norm_kernel(const int* __restrict__ src, const int* __restrict__ dst,
                            const float* __restrict__ dinv, float* __restrict__ norm, int e) {
    int t = blockIdx.x * blockDim.x + threadIdx.x;
    if (t < e) norm[t] = dinv[src[t]] * dinv[dst[t]];
}

// ---------------- GEMM: Y[rows,64] = X[rows,K] @ W[K,64] (no bias) ----------------
// One wave per 16-row tile; 4 column tiles of 16; W transposed in LDS.

template <int K>
__global__ __launch_bounds__(256) void gemm_nt64_kernel(const float* __restrict__ X,
                                                        const float* __restrict__ W,
                                                        float* __restrict__ Y,
                                                        int numTiles) {
    constexpr int KP = K + 2;  // pad to avoid LDS bank conflicts
    __shared__ float sW[64 * KP];
    const int tid = threadIdx.x;
    for (int idx = tid; idx < K * 64; idx += 256) {
        int k = idx >> 6, c = idx & 63;
        sW[c * KP + k] = W[idx];
    }
    __syncthreads();

    const int wave = tid >> 5, lane = tid & 31;
    const int tile = blockIdx.x * 8 + wave;
    if (tile >= numTiles) return;  // wave-uniform

    const int m = lane & 15, half = lane >> 4;
    const int row0 = tile * 16;
    const float* xrow = X + (size_t)(row0 + m) * K;

    v8f acc[4] = {};
#pragma unroll 4
    for (int k0 = 0; k0 < K; k0 += 4) {
        const int kl = k0 + 2 * half;
        float2 ta = *(const float2*)(xrow + kl);
        v2f a; a.x = ta.x; a.y = ta.y;
#pragma unroll
        for (int nt = 0; nt < 4; ++nt) {
            const int col = m + 16 * nt;
            float2 tb = *(const float2*)(&sW[col * KP + kl]);
            v2f b; b.x = tb.x; b.y = tb.y;
            acc[nt] = wmma_f32_k4(a, b, acc[nt]);
        }
    }
    float* yb = Y + (size_t)row0 * 64;
#pragma unroll
    for (int nt = 0; nt < 4; ++nt) {
        const int col = m + 16 * nt;
#pragma unroll
        for (int v = 0; v < 8; ++v)
            yb[(size_t)(v + 8 * half) * 64 + col] = acc[nt][v];
    }
}

// ---------------- edge scatter: agg[dst] += h[src] * norm ----------------

__global__ void scatter_kernel(const float* __restrict__ h, const int* __restrict__ src,
                               const int* __restrict__ dst, const float* __restrict__ norm,
                               float* __restrict__ agg, int e) {
    long long t = (long long)blockIdx.x * blockDim.x + threadIdx.x;
    if (t >= (long long)e * 16) return;
    int ei = (int)(t >> 4);
    int c = (int)(t & 15) * 4;
    float nv = norm[ei];
    int s = src[ei], d = dst[ei];
    float4 v = *(const float4*)(h + (size_t)s * 64 + c);
    float* ap = agg + (size_t)d * 64 + c;
    atomic_add_f32(ap + 0, v.x * nv);
    atomic_add_f32(ap + 1, v.y * nv);
    atomic_add_f32(ap + 2, v.z * nv);
    atomic_add_f32(ap + 3, v.w * nv);
}

// ---------------- bias (+ optional relu) with self-loop fold ----------------
// agg[n,c] := act(agg[n,c] + h[n,c]*dinv[n]^2 + b[c])

template <bool RELU>
__global__ void bias_self_kernel(float* __restrict__ agg, const float* __restrict__ h,
                                 const float* __restrict__ dinv, const float* __restrict__ b,
                                 int n) {
    long long t = (long long)blockIdx.x * blockDim.x + threadIdx.x;
    if (t >= (long long)n * 64) return;
    int node = (int)(t >> 6);
    int c = (int)(t & 63);
    float di = dinv[node];
    float r = agg[t] + h[t] * di * di + b[c];
    agg[t] = RELU ? fmaxf(r, 0.0f) : r;
}

// ---------------- fused MLP: out[q,:2] = relu(concat(o[i],o[j]) @ W1 + b1) @ W2 + b2 ----
// One wave per 16-query tile. fc1 via WMMA (K=128, 8 col-tiles), fc2 via VALU + shuffles.

__global__ __launch_bounds__(256) void mlp_fused_kernel(const float* __restrict__ nodeOut,
                                                        const int* __restrict__ qi,
                                                        const int* __restrict__ qj,
                                                        const float* __restrict__ W1g,
                                                        const float* __restrict__ b1g,
                                                        const float* __restrict__ W2g,
                                                        const float* __restrict__ b2g,
                                                        float* __restrict__ out,
                                                        int numTiles) {
    __shared__ float sW1[128 * 130];  // transposed [n][k], padded stride
    __shared__ float sW2[256];
    __shared__ float sB1[128];
    __shared__ float sB2[2];
    const int tid = threadIdx.x;
    for (int idx = tid; idx < 128 * 128; idx += 256) {
        int k = idx >> 7, nn = idx & 127;
        sW1[nn * 130 + k] = W1g[idx];
    }
    if (tid < 256) sW2[tid] = W2g[tid];
    if (tid < 128) sB1[tid] = b1g[tid];
    if (tid < 2)   sB2[tid] = b2g[tid];
    __syncthreads();

    const int wave = tid >> 5, lane = tid & 31;
    const int tile = blockIdx.x * 8 + wave;
    if (tile >= numTiles) return;  // wave-uniform

    const int m = lane & 15, half = lane >> 4;
    const int q = tile * 16 + m;
    const float* rowI = nodeOut + (size_t)qi[q] * 64;
    const float* rowJ = nodeOut + (size_t)qj[q] * 64;

    v8f acc[8] = {};
#pragma unroll 4
    for (int k0 = 0; k0 < 128; k0 += 4) {
        const int kl = k0 + 2 * half;
        const float* srcp = (k0 < 64) ? (rowI + kl) : (rowJ + (kl - 64));  // uniform per step
        float2 ta = *(const float2*)srcp;
        v2f a; a.x = ta.x; a.y = ta.y;
#pragma unroll
        for (int nt = 0; nt < 8; ++nt) {
            const int col = m + 16 * nt;
            float2 tb = *(const float2*)(&sW1[col * 130 + kl]);
            v2f b; b.x = tb.x; b.y = tb.y;
            acc[nt] = wmma_f32_k4(a, b, acc[nt]);
        }
    }

    // bias + relu in C-layout, then fc2 partial sums (2 output cols)
    float p0[8], p1[8];
#pragma unroll
    for (int v = 0; v < 8; ++v) { p0[v] = 0.0f; p1[v] = 0.0f; }
#pragma unroll
    for (int nt = 0; nt < 8; ++nt) {
        const int col = m + 16 * nt;  // hidden index k for fc2
        const float bv = sB1[col];
        const float w0 = sW2[col * 2 + 0];
        const float w1 = sW2[col * 2 + 1];
#pragma unroll
        for (int v = 0; v < 8; ++v) {
            float hv = fmaxf(acc[nt][v] + bv, 0.0f);
            p0[v] = fmaf(hv, w0, p0[v]);
            p1[v] = fmaf(hv, w1, p1[v]);
        }
    }
    // reduce across the 16 lanes holding the same rows (xor 8,4,2,1 stays in-half)
#pragma unroll
    for (int v = 0; v < 8; ++v) {
        for (int off = 8; off >= 1; off >>= 1) {
            p0[v] += __shfl_xor(p0[v], off);
            p1[v] += __shfl_xor(p1[v], off);
        }
    }
    if ((lane & 15) == 0) {
        const float b20 = sB2[0], b21 = sB2[1];
        const int qb = tile * 16 + 8 * half;
#pragma unroll
        for (int v = 0; v < 8; ++v) {
            out[(size_t)(qb + v) * 2 + 0] = p0[v] + b20;
            out[(size_t)(qb + v) * 2 + 1] = p1[v] + b21;
        }
    }
}

// ---------------- host ----------------

extern "C" void kernel_launch(void* const* d_in, const int* in_sizes, int n_in,
                              void* d_out, int out_size, void* d_ws, size_t ws_size,
                              hipStream_t stream) {
    const float* feature = (const float*)d_in[0];
    const int*   edges   = (const int*)d_in[1];
    const int*   qi      = (const int*)d_in[2];
    const int*   qj      = (const int*)d_in[3];
    const float* W1      = (const float*)d_in[4];
    const float* b1      = (const float*)d_in[5];
    const float* W2      = (const float*)d_in[6];
    const float* b2      = (const float*)d_in[7];
    const float* fc1_W   = (const float*)d_in[8];
    const float* fc1_b   = (const float*)d_in[9];
    const float* fc2_W   = (const float*)d_in[10];
    const float* fc2_b   = (const float*)d_in[11];
    float* dout = (float*)d_out;

    const int N = in_sizes[0] / 128;
    const int E = in_sizes[1] / 2;
    const int Q = in_sizes[2];
    const int* src = edges;
    const int* dst = edges + E;

    // workspace carve-up (all sizes multiple of 16B at these shapes)
    char* ws = (char*)d_ws;
    auto align256 = [](size_t x) { return (x + 255) & ~(size_t)255; };
    size_t off = 0;
    float* dinv = (float*)(ws + off); off = align256(off + (size_t)N * 4);
    float* normb = (float*)(ws + off); off = align256(off + (size_t)E * 4);
    float* hbuf = (float*)(ws + off); off = align256(off + (size_t)N * 64 * 4);
    float* aggbuf = (float*)(ws + off); off = align256(off + (size_t)N * 64 * 4);
    (void)ws_size;

    const int B = 256;
    // degrees + norms
    deg_init_kernel<<<(N + B - 1) / B, B, 0, stream>>>(dinv, N);
    deg_accum_kernel<<<(E + B - 1) / B, B, 0, stream>>>(dst, dinv, E);
    rsqrt_kernel<<<(N + B - 1) / B, B, 0, stream>>>(dinv, N);
    norm_kernel<<<(E + B - 1) / B, B, 0, stream>>>(src, dst, dinv, normb, E);

    const int rowTiles = (N + 15) / 16;
    const int gemmBlocks = (rowTiles + 7) / 8;
    const long long scatterThreads = (long long)E * 16;
    const unsigned scatterBlocks = (unsigned)((scatterThreads + B - 1) / B);
    const long long nodeElems = (long long)N * 64;
    const unsigned elemBlocks = (unsigned)((nodeElems + B - 1) / B);

    // layer 1: h = X @ W1 ; agg = scatter(h*norm) ; h1 = relu(agg + self + b1) (in-place in agg)
    gemm_nt64_kernel<128><<<gemmBlocks, B, 0, stream>>>(feature, W1, hbuf, rowTiles);
    hipMemsetAsync(aggbuf, 0, (size_t)N * 64 * 4, stream);
    scatter_kernel<<<scatterBlocks, B, 0, stream>>>(hbuf, src, dst, normb, aggbuf, E);
    bias_self_kernel<true><<<elemBlocks, B, 0, stream>>>(aggbuf, hbuf, dinv, b1, N);

    // layer 2: h2 = h1 @ W2 (into hbuf) ; agg = scatter(h2*norm) ; out_nodes = agg + self + b2
    gemm_nt64_kernel<64><<<gemmBlocks, B, 0, stream>>>(aggbuf, W2, hbuf, rowTiles);
    hipMemsetAsync(aggbuf, 0, (size_t)N * 64 * 4, stream);
    scatter_kernel<<<scatterBlocks, B, 0, stream>>>(hbuf, src, dst, normb, aggbuf, E);
    bias_self_kernel<false><<<elemBlocks, B, 0, stream>>>(aggbuf, hbuf, dinv, b2, N);

    // fused MLP over 1M queries
    const int qTiles = (Q + 15) / 16;
    mlp_fused_kernel<<<(qTiles + 7) / 8, B, 0, stream>>>(aggbuf, qi, qj, fc1_W, fc1_b,
                                                         fc2_W, fc2_b, dout, qTiles);
}